// SGC_IMDB_24163486007672
// MI455X (gfx1250) — compile-verified
//
#include <hip/hip_runtime.h>

typedef __attribute__((ext_vector_type(2))) float v2f;
typedef __attribute__((ext_vector_type(8))) float v8f;

#define EPS_BN 1e-5f
#define RT 4  // row tiles per wave in the bulk WMMA GEMM

static inline unsigned cdiv(long long a, long long b) { return (unsigned)((a + b - 1) / b); }
static inline size_t align_up(size_t x, size_t a) { return (x + a - 1) & ~(a - 1); }

// ---------------- generic zero ----------------
__global__ void k_zero(float* __restrict__ p, long long n) {
  long long i = (long long)blockIdx.x * blockDim.x + threadIdx.x;
  long long st = (long long)gridDim.x * blockDim.x;
  for (; i < n; i += st) p[i] = 0.0f;
}

// ---------------- degree / dnorm ----------------
__global__ void k_deg(const int* __restrict__ dst, float* __restrict__ deg, int E) {
  int i = blockIdx.x * blockDim.x + threadIdx.x;
  if (i < E) atomicAdd(&deg[dst[i]], 1.0f);
}

__global__ void k_dnorm(float* __restrict__ deg, int N) {
  int i = blockIdx.x * blockDim.x + threadIdx.x;
  if (i < N) deg[i] = rsqrtf(fmaxf(deg[i], 1.0f));
}

// ---- round-1 edge scatter: out[dst] += h[src]*dnorm[src] (h = raw node features) ----
__global__ void __launch_bounds__(256) k_scatter(const float* __restrict__ h,
                                                 const int* __restrict__ src,
                                                 const int* __restrict__ dst,
                                                 const float* __restrict__ dnorm,
                                                 float* __restrict__ out, int E, int D) {
  const int chunks = D >> 2;
  long long tid = (long long)blockIdx.x * blockDim.x + threadIdx.x;
  if (tid >= (long long)E * chunks) return;
  int e = (int)(tid / chunks);
  int c = ((int)(tid % chunks)) << 2;
  int s = src[e];
  int d = dst[e];
  float w = dnorm[s];
  const float4 v = *reinterpret_cast<const float4*>(h + (long long)s * D + c);
  float* o = out + (long long)d * D + c;
  atomicAdd(o + 0, v.x * w);
  atomicAdd(o + 1, v.y * w);
  atomicAdd(o + 2, v.z * w);
  atomicAdd(o + 3, v.w * w);
}

// ---- round-2 edge scatter with fused BN1:
//      src value = BN1(h[s]*dnorm[s]) = h[s]*dnorm[s]*sc[c] + sh[c]; message *= dnorm[s]
__global__ void __launch_bounds__(256) k_scatter_bn(const float* __restrict__ h,
                                                    const int* __restrict__ src,
                                                    const int* __restrict__ dst,
                                                    const float* __restrict__ dnorm,
                                                    const float* __restrict__ sc,
                                                    const float* __restrict__ sh,
                                                    float* __restrict__ out, int E, int D) {
  const int chunks = D >> 2;
  long long tid = (long long)blockIdx.x * blockDim.x + threadIdx.x;
  if (tid >= (long long)E * chunks) return;
  int e = (int)(tid / chunks);
  int c = ((int)(tid % chunks)) << 2;
  int s = src[e];
  int d = dst[e];
  float w = dnorm[s];
  const float4 v = *reinterpret_cast<const float4*>(h + (long long)s * D + c);
  const float4 scv = *reinterpret_cast<const float4*>(sc + c);
  const float4 shv = *reinterpret_cast<const float4*>(sh + c);
  float* o = out + (long long)d * D + c;
  atomicAdd(o + 0, (v.x * w * scv.x + shv.x) * w);
  atomicAdd(o + 1, (v.y * w * scv.y + shv.y) * w);
  atomicAdd(o + 2, (v.z * w * scv.z + shv.z) * w);
  atomicAdd(o + 3, (v.w * w * scv.w + shv.w) * w);
}

// ---- column sums / sums of squares of (h[r,:] * rowscale[r]) ----
__global__ void __launch_bounds__(256) k_colstats(const float* __restrict__ h,
                                                  const float* __restrict__ rowscale,
                                                  float* __restrict__ colsum,
                                                  float* __restrict__ colsq,
                                                  int N, int D, int use_rowscale) {
  long long gid = (long long)blockIdx.x * blockDim.x + threadIdx.x;
  int c = (int)(gid % D);
  long long r = gid / D;
  long long stride = ((long long)gridDim.x * blockDim.x) / D;
  float s = 0.0f, q = 0.0f;
  for (; r < N; r += stride) {
    float w = use_rowscale ? rowscale[r] : 1.0f;
    float v = h[r * (long long)D + c] * w;
    s += v;
    q += v * v;
  }
  atomicAdd(&colsum[c], s);
  atomicAdd(&colsq[c], q);
}

// ---- fold BN into per-column scale/shift (in-place over colsum/colsq) ----
__global__ void k_bn_prep(float* __restrict__ colsum, float* __restrict__ colsq,
                          const float* __restrict__ gamma, const float* __restrict__ beta,
                          int N, int D) {
  int c = blockIdx.x * blockDim.x + threadIdx.x;
  if (c >= D) return;
  float invN = 1.0f / (float)N;
  float mu = colsum[c] * invN;
  float var = colsq[c] * invN - mu * mu;
  float scv = rsqrtf(var + EPS_BN) * gamma[c];
  colsum[c] = scv;                  // scale
  colsq[c] = beta[c] - mu * scv;    // shift
}

// ---- GEMM: Out = BN1(A*dnorm_row) @ W + bias, fp32 WMMA 16x16x4 ----
// TILES is compile-time: straight-line inner loop (no branches -> no accumulator phi
// copies), single A base pointer with constant offsets (global_load, clause-able),
// B tile + sc/sh reused across TILES WMMAs per k-step.
// Layouts per ISA 7.12.2 (32-bit, wave32): A 16x4 -> lane<16 K={0,1}, lane>=16 K={2,3},
// M=lane&15; B mirrored with N on lanes; C/D VGPR r -> M = r + 8*(lane>>4), N = lane&15.
template <int TILES>
__global__ void __launch_bounds__(256) k_gemm_wmma(const float* __restrict__ A,
                                                   const float* __restrict__ W,
                                                   const float* __restrict__ bias,
                                                   const float* __restrict__ dnorm,
                                                   const float* __restrict__ sc,
                                                   const float* __restrict__ sh,
                                                   float* __restrict__ Out,
                                                   long long mstart, int D) {
  const int wave = threadIdx.x >> 5;   // column tile index
  const int lane = threadIdx.x & 31;
  const int half = lane >> 4;
  const int mn = lane & 15;
  const int col0 = wave * 16;
  const long long row0 = mstart + (long long)blockIdx.x * (16 * TILES);

  const float* abase = A + (row0 + mn) * (long long)D;

  float dn[TILES];
#pragma unroll
  for (int t = 0; t < TILES; ++t) dn[t] = dnorm[row0 + t * 16 + mn];

  v8f acc[TILES];
#pragma unroll
  for (int t = 0; t < TILES; ++t) acc[t] = (v8f){};

#pragma unroll 2
  for (int k = 0; k < D; k += 4) {
    const int kb = k + 2 * half;
    v2f b;
    b.x = W[(long long)kb * D + col0 + mn];
    b.y = W[(long long)(kb + 1) * D + col0 + mn];
    const v2f scv = *reinterpret_cast<const v2f*>(sc + kb);  // kb even -> 8B aligned
    const v2f shv = *reinterpret_cast<const v2f*>(sh + kb);
#pragma unroll
    for (int t = 0; t < TILES; ++t) {
      v2f a = *reinterpret_cast<const v2f*>(abase + (long long)t * 16 * D + kb);
      a.x = a.x * dn[t] * scv.x + shv.x;
      a.y = a.y * dn[t] * scv.y + shv.y;
      acc[t] = __builtin_amdgcn_wmma_f32_16x16x4_f32(false, a, false, b, (short)0,
                                                     acc[t], false, false);
    }
  }

  const float bv = bias[col0 + mn];
#pragma unroll
  for (int t = 0; t < TILES; ++t) {
    long long mbase = row0 + t * 16 + 8 * half;
#pragma unroll
    for (int r = 0; r < 8; ++r)
      Out[(mbase + r) * (long long)D + col0 + mn] = acc[t][r] + bv;
  }
}

// scalar tail for M % 16 leftover rows (same fused input transform)
__global__ void k_gemm_tail(const float* __restrict__ A, const float* __restrict__ W,
                            const float* __restrict__ bias, const float* __restrict__ dnorm,
                            const float* __restrict__ sc, const float* __restrict__ sh,
                            float* __restrict__ Out, int mstart, int M, int D) {
  int tid = blockIdx.x * blockDim.x + threadIdx.x;
  int rows = M - mstart;
  if (tid >= rows * D) return;
  int r = mstart + tid / D;
  int c = tid % D;
  float dn = dnorm[r];
  float s = bias[c];
  for (int k = 0; k < D; ++k) {
    float a = A[(long long)r * D + k] * dn * sc[k] + sh[k];
    s += a * W[(long long)k * D + c];
  }
  Out[(long long)r * D + c] = s;
}

// ---- per-graph pooling with fused BN2 + ReLU ----
__global__ void __launch_bounds__(256) k_pool_bn(const float* __restrict__ h,
                                                 const int* __restrict__ n2g,
                                                 const float* __restrict__ sc,
                                                 const float* __restrict__ sh,
                                                 float* __restrict__ pooled,
                                                 float* __restrict__ cnt, int N, int D) {
  const int chunks = D >> 2;
  long long tid = (long long)blockIdx.x * blockDim.x + threadIdx.x;
  if (tid >= (long long)N * chunks) return;
  long long i = tid / chunks;
  int c = ((int)(tid % chunks)) << 2;
  int g = n2g[i];
  float4 v = *reinterpret_cast<const float4*>(h + i * (long long)D + c);
  const float4 scv = *reinterpret_cast<const float4*>(sc + c);
  const float4 shv = *reinterpret_cast<const float4*>(sh + c);
  v.x = fmaxf(v.x * scv.x + shv.x, 0.0f);
  v.y = fmaxf(v.y * scv.y + shv.y, 0.0f);
  v.z = fmaxf(v.z * scv.z + shv.z, 0.0f);
  v.w = fmaxf(v.w * scv.w + shv.w, 0.0f);
  float* o = pooled + (long long)g * D + c;
  atomicAdd(o + 0, v.x);
  atomicAdd(o + 1, v.y);
  atomicAdd(o + 2, v.z);
  atomicAdd(o + 3, v.w);
  if (c == 0) atomicAdd(&cnt[g], 1.0f);
}

// ---- final projection: out[g,o] = mean-pooled @ pred_w + pred_b ----
__global__ void k_predict(const float* __restrict__ pooled, const float* __restrict__ cnt,
                          const float* __restrict__ pw, const float* __restrict__ pb,
                          float* __restrict__ out, int G, int D, int OUTD) {
  int tid = blockIdx.x * blockDim.x + threadIdx.x;
  if (tid >= G * OUTD) return;
  int g = tid / OUTD;
  int o = tid % OUTD;
  float inv = 1.0f / fmaxf(cnt[g], 1.0f);
  float s = 0.0f;
  for (int d = 0; d < D; ++d) s += pooled[(long long)g * D + d] * pw[d * OUTD + o];
  out[tid] = s * inv + pb[o];
}

extern "C" void kernel_launch(void* const* d_in, const int* in_sizes, int n_in,
                              void* d_out, int out_size, void* d_ws, size_t ws_size,
                              hipStream_t stream) {
  const float* nfeat  = (const float*)d_in[0];
  const int*   src    = (const int*)d_in[1];
  const int*   dst    = (const int*)d_in[2];
  const int*   n2g    = (const int*)d_in[3];
  // d_in[4] = num_graphs device scalar (G derived host-side from out_size)
  const float* conv_w = (const float*)d_in[5];
  const float* conv_b = (const float*)d_in[6];
  const float* gamma1 = (const float*)d_in[7];
  const float* beta1  = (const float*)d_in[8];
  const float* gamma2 = (const float*)d_in[9];
  const float* beta2  = (const float*)d_in[10];
  const float* pred_w = (const float*)d_in[11];
  const float* pred_b = (const float*)d_in[12];
  float* out = (float*)d_out;

  const int E    = in_sizes[1];
  const int N    = in_sizes[3];
  const int D    = in_sizes[7];    // 128
  const int OUTD = in_sizes[12];   // 2
  const int G    = out_size / OUTD;

  // ---- workspace carve-out (256B aligned regions) ----
  char* base = (char*)d_ws;
  size_t off = 0;
  auto carve = [&](size_t nfloats) {
    float* p = (float*)(base + off);
    off = align_up(off + nfloats * sizeof(float), 256);
    return p;
  };
  float* dnorm  = carve((size_t)N);
  float* bufA   = carve((size_t)N * D);
  float* bufB   = carve((size_t)N * D);
  float* colsum = carve((size_t)D);   // becomes BN scale after k_bn_prep
  float* colsq  = carve((size_t)D);   // becomes BN shift after k_bn_prep
  float* pooled = carve((size_t)G * D);
  float* cnt    = carve((size_t)G);
  (void)ws_size;

  const int BLK = 256;
  const int chunks = D >> 2;
  const long long edgeChunks = (long long)E * chunks;
  const long long elemChunks = (long long)N * chunks;

  // ---- degree + dnorm ----
  k_zero<<<cdiv(N, BLK), BLK, 0, stream>>>(dnorm, (long long)N);
  k_deg<<<cdiv(E, BLK), BLK, 0, stream>>>(dst, dnorm, E);
  k_dnorm<<<cdiv(N, BLK), BLK, 0, stream>>>(dnorm, N);

  // ---- hop 1: scatter raw features; stats of (bufA * dnorm); fold BN1 -> sc/sh ----
  k_zero<<<cdiv((long long)N * D, BLK * 8), BLK, 0, stream>>>(bufA, (long long)N * D);
  k_scatter<<<cdiv(edgeChunks, BLK), BLK, 0, stream>>>(nfeat, src, dst, dnorm, bufA, E, D);
  k_zero<<<1, BLK, 0, stream>>>(colsum, (long long)(2 * D));  // colsum+colsq contiguous
  k_colstats<<<1024, BLK, 0, stream>>>(bufA, dnorm, colsum, colsq, N, D, 1);
  k_bn_prep<<<cdiv(D, BLK), BLK, 0, stream>>>(colsum, colsq, gamma1, beta1, N, D);

  // ---- hop 2: scatter with fused BN1-of-hop1; stats of (bufB * dnorm); BN1 -> sc/sh ----
  k_zero<<<cdiv((long long)N * D, BLK * 8), BLK, 0, stream>>>(bufB, (long long)N * D);
  k_scatter_bn<<<cdiv(edgeChunks, BLK), BLK, 0, stream>>>(bufA, src, dst, dnorm,
                                                          colsum, colsq, bufB, E, D);
  k_zero<<<1, BLK, 0, stream>>>(colsum, (long long)(2 * D));
  k_colstats<<<1024, BLK, 0, stream>>>(bufB, dnorm, colsum, colsq, N, D, 1);
  k_bn_prep<<<cdiv(D, BLK), BLK, 0, stream>>>(colsum, colsq, gamma1, beta1, N, D);

  // ---- SGConv linear via fp32 WMMA with fused BN1(hop2) on the A operand ----
  float* h2 = bufA;  // reuse
  const int mtiles = N / 16;
  const int bulk = mtiles / RT;        // blocks of RT full row-tiles
  const int remt = mtiles % RT;        // leftover full row-tiles
  const int wavesPerBlk = 32 * (D / 16);
  if (bulk > 0)
    k_gemm_wmma<RT><<<bulk, wavesPerBlk, 0, stream>>>(bufB, conv_w, conv_b, dnorm,
                                                      colsum, colsq, h2, 0LL, D);
  if (remt > 0)
    k_gemm_wmma<1><<<remt, wavesPerBlk, 0, stream>>>(bufB, conv_w, conv_b, dnorm,
                                                     colsum, colsq, h2,
                                                     (long long)bulk * RT * 16, D);
  const int mrem = N - mtiles * 16;
  if (mrem > 0)
    k_gemm_tail<<<cdiv((long long)mrem * D, BLK), BLK, 0, stream>>>(
        bufB, conv_w, conv_b, dnorm, colsum, colsq, h2, mtiles * 16, N, D);

  // ---- BN2 stats, then pooling with fused BN2 + ReLU ----
  k_zero<<<1, BLK, 0, stream>>>(colsum, (long long)(2 * D));
  k_colstats<<<1024, BLK, 0, stream>>>(h2, dnorm, colsum, colsq, N, D, 0);
  k_bn_prep<<<cdiv(D, BLK), BLK, 0, stream>>>(colsum, colsq, gamma2, beta2, N, D);

  k_zero<<<cdiv((long long)G * D, BLK), BLK, 0, stream>>>(pooled, (long long)G * D);
  k_zero<<<cdiv(G, BLK), BLK, 0, stream>>>(cnt, (long long)G);
  k_pool_bn<<<cdiv(elemChunks, BLK), BLK, 0, stream>>>(h2, n2g, colsum, colsq,
                                                       pooled, cnt, N, D);

  // ---- final projection ----
  k_predict<<<cdiv((long long)G * OUTD, BLK), BLK, 0, stream>>>(pooled, cnt, pred_w,
                                                                pred_b, out, G, D, OUTD);
}